// SparseMaxPool_12438225289333
// MI455X (gfx1250) — compile-verified
//
#include <hip/hip_runtime.h>
#include <cstdint>

#define NN 64
#define TILE_ELEMS (NN * NN)   // 4096 floats = 16 KB per tile

// ---- compile-time sparsity pattern from POOLING_COUNTS = [15, 8, 8] ----
constexpr bool pm(int i, int j) {
    int d = j - i;
    if (d < 0) return false;
    if (d <= 15) return true;                               // diag + stage 0
    if (d <= 31) return ((d & 1) == 1) && ((i & 1) == 0);   // stage 1
    return ((d & 3) == 3) && ((i & 3) == 0);                // stage 2
}
struct CMtbl { unsigned long long m[NN]; };
constexpr CMtbl make_cm() {
    CMtbl t{};
    for (int j = 0; j < NN; ++j) {
        unsigned long long v = 0;
        for (int i = 0; i < NN; ++i) if (pm(i, j)) v |= (1ull << i);
        t.m[j] = v;
    }
    return t;
}
__constant__ CMtbl CMASK = make_cm();   // bit i of m[j] == mask(i, j)

// One 64-thread workgroup (2 wave32s) per 64x64 tile.
// Blocks [0,rows): map2d tiles; [rows,rows+B): mask2d tiles. Both at out + wg*4096.
__global__ __launch_bounds__(64, 8)
void sparse_maxpool_tiles(const float* __restrict__ x, float* __restrict__ out, int rows) {
    __shared__ float tile[TILE_ELEMS];

    const int j  = threadIdx.x;          // lane owns column j
    const int wg = blockIdx.x;

    const unsigned long long cm = CMASK.m[j];          // coalesced 8B/lane constant load
    const uint32_t cmlo = (uint32_t)cm;
    const uint32_t cmhi = (uint32_t)(cm >> 32);

    if (wg < rows) {
        const float* xrow = x + (size_t)wg * NN;       // uniform address -> SMEM s_load
        float xs[NN];
        #pragma unroll
        for (int t = 0; t < NN; ++t) xs[t] = xrow[t];

        float run = 0.0f;
        #pragma unroll
        for (int i = NN - 1; i >= 0; --i) {
            float xi = xs[i];
            run = (j == i) ? xi : fmaxf(run, xi);      // v_cmp_eq + v_cndmask + v_max
            uint32_t w = (i < 32) ? cmlo : cmhi;       // compile-time choice per i
            // forced v_bfe_i32: 0xFFFFFFFF if mask(i,j) else 0
            uint32_t keep = (uint32_t)__builtin_amdgcn_sbfe((int)w, i & 31, 1);
            uint32_t bits = __float_as_uint(run) & keep;        // exact run or +0.0
            tile[i * NN + j] = __uint_as_float(bits);  // lane-consecutive, conflict-free
        }
    } else {
        #pragma unroll
        for (int i = 0; i < NN; ++i) {
            uint32_t w = (i < 32) ? cmlo : cmhi;
            // forced v_bfe_u32 (0/1) + v_cvt_f32_u32 -> exact 0.0f / 1.0f
            uint32_t bit = __builtin_amdgcn_ubfe(w, i & 31, 1);
            tile[i * NN + j] = (float)bit;
        }
    }
    __syncthreads();   // tile complete workgroup-wide (s_wait_dscnt + barrier)

    // 16 x 1KB async LDS->global b128 streaming stores (ASYNCcnt).
    // SADDR form: global = SADDR + VADDR + IOFFSET, LDS = VSRC + IOFFSET,
    // so all 16 stores share the same two VGPRs + one SGPR pair.
    const uint64_t dstbase = (uint64_t)(uintptr_t)(out + (size_t)wg * TILE_ELEMS);
    const uint32_t voff    = (uint32_t)(j * 16);
    const uint32_t laddr   = ((uint32_t)(uintptr_t)&tile[0]) + voff;  // flat->LDS: addr[31:0]
    #define AST(OFF) asm volatile(                                              \
        "global_store_async_from_lds_b128 %0, %1, %2 offset:" #OFF              \
        " th:TH_STORE_NT"                                                       \
        :: "v"(voff), "v"(laddr), "s"(dstbase) : "memory")
    AST(0);     AST(1024);  AST(2048);  AST(3072);
    AST(4096);  AST(5120);  AST(6144);  AST(7168);
    AST(8192);  AST(9216);  AST(10240); AST(11264);
    AST(12288); AST(13312); AST(14336); AST(15360);
    #undef AST
    // don't retire the wave/LDS before the async engine has drained the tile
    asm volatile("s_wait_asynccnt 0" ::: "memory");
}

extern "C" void kernel_launch(void* const* d_in, const int* in_sizes, int n_in,
                              void* d_out, int out_size, void* d_ws, size_t ws_size,
                              hipStream_t stream) {
    (void)n_in; (void)d_ws; (void)ws_size; (void)out_size;
    const float* x = (const float*)d_in[0];
    float* out = (float*)d_out;

    const int rows = in_sizes[0] / NN;   // B*D = 16384
    const int B    = rows / 512;         // 32 mask tiles

    dim3 grid((unsigned)(rows + B));
    dim3 block(64);
    hipLaunchKernelGGL(sparse_maxpool_tiles, grid, block, 0, stream, x, out, rows);
}